// RNAFeatures_32272384262604
// MI455X (gfx1250) — compile-verified
//
#include <hip/hip_runtime.h>
#include <hip/hip_bf16.h>
#include <math.h>

// ---------------- problem constants ----------------
constexpr int BB = 8;
constexpr int NNODE = 1024;
constexpr int KNB = 30;                 // top-k neighbors
constexpr int NRBF = 16;
constexpr int NODE_IN = 101;
constexpr int EDGE_IN = 115;
constexpr int FEAT = 128;               // NODE_F == EDGE_F
constexpr int NODE_KP = 104;            // 101 padded to multiple of 4
constexpr int EDGE_KP = 116;            // 115 padded to multiple of 4
constexpr int NROWS = BB * NNODE;       // 8192
constexpr int EROWS = BB * NNODE * KNB; // 245760

// ---------------- output layout (floats) ----------------
constexpr size_t OUT_X    = 0;                                   // 147456
constexpr size_t OUT_S    = OUT_X + (size_t)NROWS * 18;          // 8192
constexpr size_t OUT_HV   = OUT_S + NROWS;                       // 8192*128
constexpr size_t OUT_HE   = OUT_HV + (size_t)NROWS * FEAT;       // 245760*128
constexpr size_t OUT_EIDX = OUT_HE + (size_t)EROWS * FEAT;       // 2*245760
constexpr size_t OUT_BID  = OUT_EIDX + (size_t)2 * EROWS;        // 8192

// ---------------- small math helpers ----------------
struct F3 { float x, y, z; };
__device__ __forceinline__ F3 f3(float a, float b, float c) { F3 r; r.x=a; r.y=b; r.z=c; return r; }
__device__ __forceinline__ F3 sub3(F3 a, F3 b) { return f3(a.x-b.x, a.y-b.y, a.z-b.z); }
__device__ __forceinline__ float dot3(F3 a, F3 b) { return a.x*b.x + a.y*b.y + a.z*b.z; }
__device__ __forceinline__ F3 cross3(F3 a, F3 b) {
  return f3(a.y*b.z - a.z*b.y, a.z*b.x - a.x*b.z, a.x*b.y - a.y*b.x);
}
__device__ __forceinline__ F3 l2n(F3 a) {
  float n = sqrtf(dot3(a, a));
  float inv = 1.0f / fmaxf(n, 1e-12f);
  return f3(a.x*inv, a.y*inv, a.z*inv);
}
__device__ __forceinline__ float fsign(float x) { return (x > 0.f) ? 1.f : ((x < 0.f) ? -1.f : 0.f); }
// X layout: (B, N, 6, 3)
__device__ __forceinline__ F3 ldx(const float* X, int b, int n, int a) {
  size_t o = (((size_t)b * NNODE + (size_t)n) * 6 + (size_t)a) * 3;
  return f3(X[o], X[o+1], X[o+2]);
}
// flattened Xf within one batch (Xb points at X + b*N*18); f in [0, 6N)
__device__ __forceinline__ F3 ldf(const float* Xb, int f) {
  size_t o = (size_t)f * 3;
  return f3(Xb[o], Xb[o+1], Xb[o+2]);
}

// ---------------- misc: X copy, S cast, batch_id ----------------
__global__ void k_misc(const float* __restrict__ X, const int* __restrict__ S,
                       float* __restrict__ out) {
  int i = blockIdx.x * blockDim.x + threadIdx.x;
  int nx = NROWS * 18;
  if (i < nx) {
    out[OUT_X + i] = X[i];
  } else if (i < nx + NROWS) {
    int t = i - nx;
    out[OUT_S + t] = (float)S[t];
    out[OUT_BID + t] = (float)(t / NNODE);
  }
}

// ---------------- pad weights into K_PAD x 128 ----------------
__global__ void k_wpad(const float* __restrict__ nW, const float* __restrict__ eW,
                       float* __restrict__ Wn, float* __restrict__ We) {
  int i = blockIdx.x * blockDim.x + threadIdx.x;
  if (i < NODE_KP * FEAT) {
    int r = i / FEAT, c = i % FEAT;
    Wn[i] = (r < NODE_IN) ? nW[r * FEAT + c] : 0.f;
  } else {
    int t = i - NODE_KP * FEAT;
    if (t < EDGE_KP * FEAT) {
      int r = t / FEAT, c = t % FEAT;
      We[t] = (r < EDGE_IN) ? eW[r * FEAT + c] : 0.f;
    }
  }
}

// ---------------- batch shifts (cumsum of mask lengths) ----------------
__global__ void k_shift(const float* __restrict__ mask, int* __restrict__ shift) {
  if (blockIdx.x == 0 && threadIdx.x == 0) {
    int cum = 0;
    for (int b = 0; b < BB; ++b) {
      shift[b] = cum;
      float s = 0.f;
      for (int n = 0; n < NNODE; ++n) s += mask[(size_t)b * NNODE + n];
      cum += (int)(s + 0.5f);
    }
  }
}

// ---------------- masked pairwise top-k (stable, matches lax.top_k ties) ----------------
__global__ void k_topk(const float* __restrict__ X, const float* __restrict__ mask,
                       int* __restrict__ Eidx) {
  int idx = blockIdx.x * blockDim.x + threadIdx.x;
  if (idx >= NROWS) return;
  int b = idx / NNODE, i = idx % NNODE;
  const float* Xb = X + (size_t)b * NNODE * 18;
  float xi0 = Xb[(size_t)i*18+0], xi1 = Xb[(size_t)i*18+1], xi2 = Xb[(size_t)i*18+2];
  float mi = mask[(size_t)b * NNODE + i];
  // pass 1: Dmax
  float Dmax = -3.4e38f;
  for (int j = 0; j < NNODE; ++j) {
    float m2 = mi * mask[(size_t)b * NNODE + j];
    float dx = xi0 - Xb[(size_t)j*18+0];
    float dy = xi1 - Xb[(size_t)j*18+1];
    float dz = xi2 - Xb[(size_t)j*18+2];
    float d = sqrtf(dx*dx + dy*dy + dz*dz + 1e-6f);
    float D = (1.f - m2) * 10000.f + m2 * d;
    Dmax = fmaxf(Dmax, D);
  }
  // pass 2: stable insertion top-30 of Dadj
  float vals[KNB]; int ids[KNB];
  for (int t = 0; t < KNB; ++t) { vals[t] = 3.4e38f; ids[t] = 0; }
  for (int j = 0; j < NNODE; ++j) {
    float m2 = mi * mask[(size_t)b * NNODE + j];
    float dx = xi0 - Xb[(size_t)j*18+0];
    float dy = xi1 - Xb[(size_t)j*18+1];
    float dz = xi2 - Xb[(size_t)j*18+2];
    float d = sqrtf(dx*dx + dy*dy + dz*dz + 1e-6f);
    float D = (1.f - m2) * 10000.f + m2 * d;
    float Dadj = D + (1.f - m2) * (Dmax + 1.f);
    if (Dadj < vals[KNB-1]) {
      int p = KNB - 1;
      while (p > 0 && Dadj < vals[p-1]) { vals[p] = vals[p-1]; ids[p] = ids[p-1]; --p; }
      vals[p] = Dadj; ids[p] = j;
    }
  }
  for (int t = 0; t < KNB; ++t) Eidx[(size_t)idx * KNB + t] = ids[t];
}

// ---------------- E_idx output (dst; src) as floats ----------------
__global__ void k_eidxout(const int* __restrict__ Eidx, const int* __restrict__ shift,
                          float* __restrict__ out) {
  int e = blockIdx.x * blockDim.x + threadIdx.x;
  if (e >= EROWS) return;
  int b = e / (NNODE * KNB);
  int n = (e / KNB) % NNODE;
  int s = shift[b];
  out[OUT_EIDX + e] = (float)(s + n);
  out[OUT_EIDX + (size_t)EROWS + e] = (float)(s + Eidx[e]);
}

// ---------------- node features part A: dihedrals (12) + node RBFs (80) ----------------
__global__ void k_nodeA(const float* __restrict__ X, float* __restrict__ nodeF) {
  int idx = blockIdx.x * blockDim.x + threadIdx.x;
  if (idx >= NROWS) return;
  int b = idx / NNODE, n = idx % NNODE;
  const float* Xb = X + (size_t)b * NNODE * 18;
  float* out = nodeF + (size_t)idx * NODE_KP;
  const int FN = 6 * NNODE;
  // dihedrals over the flattened chain, lag-5 differencing (per reference)
  for (int o = 0; o < 6; ++o) {
    int kx = 6 * n + o;
    float Dv = 0.f;
    if (kx >= 3 && kx <= FN - 5) {
      int t = kx - 3;
      F3 u2 = l2n(sub3(ldf(Xb, t + 5), ldf(Xb, t)));
      F3 u1 = l2n(sub3(ldf(Xb, t + 6), ldf(Xb, t + 1)));
      F3 u0 = l2n(sub3(ldf(Xb, t + 7), ldf(Xb, t + 2)));
      F3 n2 = l2n(cross3(u2, u1));
      F3 n1 = l2n(cross3(u1, u0));
      float cd = fminf(fmaxf(dot3(n2, n1), -1.f + 1e-7f), 1.f - 1e-7f);
      Dv = fsign(dot3(u2, n1)) * acosf(cd);
    }
    out[o]     = cosf(Dv);
    out[6 + o] = sinf(Dv);
  }
  // node-pair RBFs: atoms (1..5) vs atom 0
  F3 p0 = ldx(X, b, n, 0);
  for (int p = 0; p < 5; ++p) {
    F3 d = sub3(ldx(X, b, n, p + 1), p0);
    float dist = sqrtf(dot3(d, d) + 1e-6f);
    for (int r = 0; r < NRBF; ++r) {
      float mu = 20.f * (float)r / 15.f;
      float z = (dist - mu) * (1.f / 1.25f);
      out[12 + p * NRBF + r] = expf(-z * z);
    }
  }
}

// ---------------- node features part B: local frames Q + V_direct (9) ----------------
__global__ void k_nodeB(const float* __restrict__ X, float* __restrict__ Q,
                        float* __restrict__ nodeF) {
  int idx = blockIdx.x * blockDim.x + threadIdx.x;
  if (idx >= NROWS) return;
  int b = idx / NNODE, n = idx % NNODE;
  float q[9];
  if (n == NNODE - 1) {
    for (int t = 0; t < 9; ++t) q[t] = 0.f;
  } else {
    F3 p4 = ldx(X, b, n, 4), p5 = ldx(X, b, n, 5), r0 = ldx(X, b, n + 1, 0);
    F3 u0 = l2n(sub3(p5, p4));
    F3 u1 = l2n(sub3(r0, p5));
    F3 n0v = l2n(cross3(u0, u1));
    F3 b1  = l2n(sub3(u0, u1));
    F3 c2  = cross3(b1, n0v);
    q[0]=b1.x; q[1]=b1.y; q[2]=b1.z;
    q[3]=n0v.x; q[4]=n0v.y; q[5]=n0v.z;
    q[6]=c2.x; q[7]=c2.y; q[8]=c2.z;
  }
  for (int t = 0; t < 9; ++t) Q[(size_t)idx * 9 + t] = q[t];
  // V_direct: own atoms {0,2,3} relative to Xc (atom 4), rotated by Qm, normalized
  F3 Xc = ldx(X, b, n, 4);
  const int alist[3] = {0, 2, 3};
  float* out = nodeF + (size_t)idx * NODE_KP;
  for (int ai = 0; ai < 3; ++ai) {
    F3 dx = sub3(ldx(X, b, n, alist[ai]), Xc);
    F3 du = f3(q[0]*dx.x + q[1]*dx.y + q[2]*dx.z,
               q[3]*dx.x + q[4]*dx.y + q[5]*dx.z,
               q[6]*dx.x + q[7]*dx.y + q[8]*dx.z);
    du = l2n(du);
    out[92 + ai*3 + 0] = du.x;
    out[92 + ai*3 + 1] = du.y;
    out[92 + ai*3 + 2] = du.z;
  }
  out[101] = 0.f; out[102] = 0.f; out[103] = 0.f;  // K padding
}

// ---------------- edge features: quaternion(4) + edge RBFs(96) + E_direct(15) ----------------
__global__ void k_edgef(const float* __restrict__ X, const float* __restrict__ Q,
                        const int* __restrict__ Eidx, float* __restrict__ edgeF) {
  int e = blockIdx.x * blockDim.x + threadIdx.x;
  if (e >= EROWS) return;
  int b = e / (NNODE * KNB);
  int n = (e / KNB) % NNODE;
  int j = Eidx[e];
  const float* qm = Q + ((size_t)b * NNODE + n) * 9;
  const float* qn = Q + ((size_t)b * NNODE + j) * 9;
  float* out = edgeF + (size_t)e * EDGE_KP;
  // R[x][z] = sum_y qm[y][x] * qn[y][z]  (Qm^T Qn)
  float R[3][3];
  for (int x = 0; x < 3; ++x)
    for (int z = 0; z < 3; ++z)
      R[x][z] = qm[0+x]*qn[0+z] + qm[3+x]*qn[3+z] + qm[6+x]*qn[6+z];
  float Rxx = R[0][0], Ryy = R[1][1], Rzz = R[2][2];
  float m0 = 0.5f * sqrtf(fabsf(1.f + Rxx - Ryy - Rzz));
  float m1 = 0.5f * sqrtf(fabsf(1.f - Rxx + Ryy - Rzz));
  float m2 = 0.5f * sqrtf(fabsf(1.f - Rxx - Ryy + Rzz));
  float qx = fsign(R[2][1] - R[1][2]) * m0;
  float qy = fsign(R[0][2] - R[2][0]) * m1;
  float qz = fsign(R[1][0] - R[0][1]) * m2;
  float qw = 0.5f * sqrtf(fmaxf(0.f, 1.f + Rxx + Ryy + Rzz));
  float qlen = sqrtf(qx*qx + qy*qy + qz*qz + qw*qw);
  float inv = 1.f / fmaxf(qlen, 1e-12f);
  out[0] = qx*inv; out[1] = qy*inv; out[2] = qz*inv; out[3] = qw*inv;
  // edge RBFs: own atoms 0..5 vs neighbor atom 0
  F3 pj0 = ldx(X, b, j, 0);
  for (int a = 0; a < 6; ++a) {
    F3 d = sub3(ldx(X, b, n, a), pj0);
    float dist = sqrtf(dot3(d, d) + 1e-6f);
    for (int r = 0; r < NRBF; ++r) {
      float mu = 20.f * (float)r / 15.f;
      float z = (dist - mu) * (1.f / 1.25f);
      out[4 + a * NRBF + r] = expf(-z * z);
    }
  }
  // E_direct: neighbor atoms {0,1,2,3,5} relative to own Xc, rotated by Qm
  F3 Xc = ldx(X, b, n, 4);
  const int alist[5] = {0, 1, 2, 3, 5};
  for (int ai = 0; ai < 5; ++ai) {
    F3 dx = sub3(ldx(X, b, j, alist[ai]), Xc);
    F3 du = f3(qm[0]*dx.x + qm[1]*dx.y + qm[2]*dx.z,
               qm[3]*dx.x + qm[4]*dx.y + qm[5]*dx.z,
               qm[6]*dx.x + qm[7]*dx.y + qm[8]*dx.z);
    du = l2n(du);
    out[100 + ai*3 + 0] = du.x;
    out[100 + ai*3 + 1] = du.y;
    out[100 + ai*3 + 2] = du.z;
  }
  out[115] = 0.f;  // K padding
}

// ---------------- fp32 WMMA GEMM: Out[M,128] = F[M,Kpad] @ Wp[Kpad,128] + bias ----------------
typedef __attribute__((ext_vector_type(2))) float v2f;
typedef __attribute__((ext_vector_type(8))) float v8f;

__global__ void k_gemm(const float* __restrict__ F, const float* __restrict__ Wp,
                       const float* __restrict__ bias, float* __restrict__ Out,
                       int Kpad) {
  // 256 threads = 8 waves; wave w owns 16-column tile n0 = w*16; 4 row-tiles of 16.
  int wave = threadIdx.x >> 5;
  int lane = threadIdx.x & 31;
  int lh = lane & 15;           // lane within half
  int khalf = lane >> 4;        // 0: K pair {0,1}; 1: K pair {2,3} (A layout, ISA 7.12.2)
  int n0 = wave * 16;
  int rowBase = blockIdx.x * 64;
  for (int rt = 0; rt < 4; ++rt) {
    int row0 = rowBase + rt * 16;
    v8f c = {};
    const float* frow = F + (size_t)(row0 + lh) * Kpad + khalf * 2;
    const float* wcol = Wp + (size_t)(khalf * 2) * FEAT + n0 + lh;
    for (int k0 = 0; k0 < Kpad; k0 += 4) {
      v2f a; a.x = frow[k0];                 a.y = frow[k0 + 1];
      v2f bm; bm.x = wcol[(size_t)k0 * FEAT]; bm.y = wcol[(size_t)(k0 + 1) * FEAT];
      c = __builtin_amdgcn_wmma_f32_16x16x4_f32(false, a, false, bm,
                                                (short)0, c, false, false);
    }
    // D layout: VGPR r -> (m = r + khalf*8, n = n0 + lh); fuse bias add.
    int mbase = row0 + khalf * 8;
    int col = n0 + lh;
    float bc = bias[col];
#pragma unroll
    for (int r = 0; r < 8; ++r)
      Out[(size_t)(mbase + r) * FEAT + col] = c[r] + bc;
  }
}

// ---------------- layernorm (ddof=1, double-eps), one wave per 128-wide row ----------------
__global__ void k_ln(float* __restrict__ Out, const float* __restrict__ gain,
                     const float* __restrict__ bias, int rows) {
  int row = blockIdx.x * 8 + (threadIdx.x >> 5);
  int lane = threadIdx.x & 31;
  if (row >= rows) return;
  float* rp = Out + (size_t)row * FEAT;
  float4 x = reinterpret_cast<const float4*>(rp)[lane];
  float s = x.x + x.y + x.z + x.w;
#pragma unroll
  for (int m = 16; m >= 1; m >>= 1) s += __shfl_xor(s, m, 32);
  float mu = s * (1.f / 128.f);
  float d0 = x.x - mu, d1 = x.y - mu, d2 = x.z - mu, d3 = x.w - mu;
  float v = d0*d0 + d1*d1 + d2*d2 + d3*d3;
#pragma unroll
  for (int m = 16; m >= 1; m >>= 1) v += __shfl_xor(v, m, 32);
  float sigma = sqrtf(v * (1.f / 127.f) + 1e-6f);
  float inv = 1.f / (sigma + 1e-6f);
  float4 g = reinterpret_cast<const float4*>(gain)[lane];
  float4 bb = reinterpret_cast<const float4*>(bias)[lane];
  float4 y;
  y.x = g.x * d0 * inv + bb.x;
  y.y = g.y * d1 * inv + bb.y;
  y.z = g.z * d2 * inv + bb.z;
  y.w = g.w * d3 * inv + bb.w;
  reinterpret_cast<float4*>(rp)[lane] = y;
}

// ---------------- host launch ----------------
extern "C" void kernel_launch(void* const* d_in, const int* in_sizes, int n_in,
                              void* d_out, int out_size, void* d_ws, size_t ws_size,
                              hipStream_t stream) {
  (void)in_sizes; (void)n_in; (void)out_size; (void)ws_size;
  const float* X      = (const float*)d_in[0];
  const int*   S      = (const int*)d_in[1];
  const float* mask   = (const float*)d_in[2];
  const float* node_W = (const float*)d_in[3];
  const float* node_b = (const float*)d_in[4];
  const float* edge_W = (const float*)d_in[5];
  const float* edge_b = (const float*)d_in[6];
  const float* ngain  = (const float*)d_in[7];
  const float* nbias  = (const float*)d_in[8];
  const float* egain  = (const float*)d_in[9];
  const float* ebias  = (const float*)d_in[10];
  float* out = (float*)d_out;

  // workspace partition (4-byte words)
  int*   ws_Eidx  = (int*)d_ws;                         // EROWS ints
  int*   ws_shift = ws_Eidx + EROWS;                    // 8 ints (padded to 64)
  float* ws_Q     = (float*)(ws_shift + 64);            // NROWS*9
  float* ws_Wn    = ws_Q + (size_t)NROWS * 9;           // NODE_KP*128
  float* ws_We    = ws_Wn + (size_t)NODE_KP * FEAT;     // EDGE_KP*128
  float* ws_nF    = ws_We + (size_t)EDGE_KP * FEAT;     // NROWS*NODE_KP
  float* ws_eF    = ws_nF + (size_t)NROWS * NODE_KP;    // EROWS*EDGE_KP (~114 MB)

  int miscN = NROWS * 18 + NROWS;
  k_misc<<<(miscN + 255) / 256, 256, 0, stream>>>(X, S, out);
  k_wpad<<<((NODE_KP + EDGE_KP) * FEAT + 255) / 256, 256, 0, stream>>>(node_W, edge_W, ws_Wn, ws_We);
  k_shift<<<1, 1, 0, stream>>>(mask, ws_shift);
  k_topk<<<NROWS / 128, 128, 0, stream>>>(X, mask, ws_Eidx);
  k_eidxout<<<(EROWS + 255) / 256, 256, 0, stream>>>(ws_Eidx, ws_shift, out);
  k_nodeA<<<NROWS / 128, 128, 0, stream>>>(X, ws_nF);
  k_nodeB<<<NROWS / 128, 128, 0, stream>>>(X, ws_Q, ws_nF);
  k_edgef<<<EROWS / 128, 128, 0, stream>>>(X, ws_Q, ws_Eidx, ws_eF);
  k_gemm<<<NROWS / 64, 256, 0, stream>>>(ws_nF, ws_Wn, node_b, out + OUT_HV, NODE_KP);
  k_gemm<<<EROWS / 64, 256, 0, stream>>>(ws_eF, ws_We, edge_b, out + OUT_HE, EDGE_KP);
  k_ln<<<NROWS / 8, 256, 0, stream>>>(out + OUT_HV, ngain, nbias, NROWS);
  k_ln<<<EROWS / 8, 256, 0, stream>>>(out + OUT_HE, egain, ebias, EROWS);
}